// GINF_51548197486839
// MI455X (gfx1250) — compile-verified
//
#include <hip/hip_runtime.h>
#include <math.h>

#define NN   100000
#define EE   1600000
#define DD   64
#define HH   64
#define MM   8
#define LL   3
#define GG   64
#define OUTC 32
#define DM   72   // D + M

typedef __attribute__((ext_vector_type(2))) float v2f;
typedef __attribute__((ext_vector_type(8))) float v8f;

// ---------------------------------------------------------------------------
// Kernel 1: random Fourier features + concat:  hcat = [h, cos(h@W + b)/sqrt(M)]
// One thread per node. W (64x8) is wave-uniform -> scalar cache loads.
// ---------------------------------------------------------------------------
__global__ __launch_bounds__(256)
void rff_concat(const float* __restrict__ h, const float* __restrict__ W,
                const float* __restrict__ b, float* __restrict__ hcat, int n) {
    int node = blockIdx.x * blockDim.x + threadIdx.x;
    if (node >= n) return;
    const float4* h4 = (const float4*)(h + (long)node * DD);
    float*        oc = hcat + (long)node * DM;
    float4*       o4 = (float4*)oc;

    float acc[MM];
#pragma unroll
    for (int m = 0; m < MM; ++m) acc[m] = b[m];

#pragma unroll
    for (int q = 0; q < DD / 4; ++q) {
        float4 v = h4[q];
        o4[q] = v;                      // pass-through copy of h into hcat[:,0:64]
        int d = q * 4;
#pragma unroll
        for (int m = 0; m < MM; ++m) {
            acc[m] = fmaf(v.x, W[(d + 0) * MM + m], acc[m]);
            acc[m] = fmaf(v.y, W[(d + 1) * MM + m], acc[m]);
            acc[m] = fmaf(v.z, W[(d + 2) * MM + m], acc[m]);
            acc[m] = fmaf(v.w, W[(d + 3) * MM + m], acc[m]);
        }
    }
    const float s = 0.35355339059327373f;   // 1/sqrt(8)
#pragma unroll
    for (int m = 0; m < MM; ++m) oc[DD + m] = s * cosf(acc[m]);
}

// ---------------------------------------------------------------------------
// Kernel 2: GIN aggregation  agg[dst] += hcat[src]  (edge-parallel, L2 atomics)
// Node array (28.8 MB) is L2-resident on MI455X (192 MB L2).
// ---------------------------------------------------------------------------
__global__ __launch_bounds__(256)
void edge_agg(const int* __restrict__ src, const int* __restrict__ dst,
              const float* __restrict__ hcat, float* __restrict__ agg, int e) {
    int t = blockIdx.x * blockDim.x + threadIdx.x;
    if (t >= e) return;
    long s = src[t];
    long d = dst[t];
    const float4* hp = (const float4*)(hcat + s * DM);
    float*        ap = agg + d * DM;
#pragma unroll
    for (int q = 0; q < DM / 4; ++q) {
        float4 v = hp[q];
        atomicAdd(ap + q * 4 + 0, v.x);
        atomicAdd(ap + q * 4 + 1, v.y);
        atomicAdd(ap + q * 4 + 2, v.z);
        atomicAdd(ap + q * 4 + 3, v.w);
    }
}

// ---------------------------------------------------------------------------
// Kernel 3: fused GIN MLP, fp32 WMMA (V_WMMA_F32_16X16X4_F32).
//   hsum = hcat + agg          [16 x 72] per-wave LDS tile (stride 74: 8B-
//                              aligned K-pairs -> single ds_load_b64, and
//                              10*r mod 64 bank permutation -> conflict-free)
//   t    = relu(hsum@w1 + b1)  written back into the consumed hs tile
//   hout = relu(t@w2 + b2)
// B operands pre-swizzled into WMMA lane order: bs[kq][nt][lane] = {w[k][col],
// w[k+1][col]} with k = 4*kq + 2*(lane>=16), col = 16*nt + (lane&15). Each
// lane's operand is one lane-linear aligned ds_load_b64 (banks 2*lane) -> no
// repacking moves, no bank conflicts.
// ---------------------------------------------------------------------------
__global__ __launch_bounds__(256)
void gin_mlp(const float* __restrict__ hcat, const float* __restrict__ agg,
             const float* __restrict__ w1, const float* __restrict__ b1,
             const float* __restrict__ w2, const float* __restrict__ b2,
             float* __restrict__ hout, int n) {
    __shared__ v2f   bs1[(DM / 4) * 4 * 32];   // 18*128 v2f = 18 KB
    __shared__ v2f   bs2[(HH / 4) * 4 * 32];   // 16*128 v2f = 16 KB
    __shared__ float b1s[HH], b2s[HH];
    __shared__ float hs[8][16][74];            // 37 KB, reused for t after GEMM1

    const int tid = threadIdx.x;

    // ---- stage weights into WMMA-ready lane order (block cooperative) ----
    for (int i = tid; i < (DM / 4) * 128; i += 256) {
        int kq    = i >> 7;
        int nt    = (i >> 5) & 3;
        int lane2 = i & 31;
        int k     = kq * 4 + ((lane2 >> 4) << 1);
        int col   = nt * 16 + (lane2 & 15);
        v2f v;
        v.x = w1[k * HH + col];
        v.y = w1[(k + 1) * HH + col];
        bs1[i] = v;
    }
    for (int i = tid; i < (HH / 4) * 128; i += 256) {
        int kq    = i >> 7;
        int nt    = (i >> 5) & 3;
        int lane2 = i & 31;
        int k     = kq * 4 + ((lane2 >> 4) << 1);
        int col   = nt * 16 + (lane2 & 15);
        v2f v;
        v.x = w2[k * HH + col];
        v.y = w2[(k + 1) * HH + col];
        bs2[i] = v;
    }
    if (tid < HH) { b1s[tid] = b1[tid]; b2s[tid] = b2[tid]; }

    const int lane  = tid & 31;
    const int wave  = tid >> 5;
    const int half  = lane >> 4;      // 0: lanes 0-15, 1: lanes 16-31
    const int l16   = lane & 15;
    const int node0 = blockIdx.x * 128 + wave * 16;

    // ---- stage hsum = hcat + agg tile (wave-private region of LDS) ----
    for (int idx = lane; idx < 16 * (DM / 4); idx += 32) {
        int r  = idx / (DM / 4);
        int c4 = idx % (DM / 4);
        int node = node0 + r;
        if (node >= n) node = n - 1;            // clamp tail (stores guarded later)
        float4 hv = ((const float4*)(hcat + (long)node * DM))[c4];
        float4 av = ((const float4*)(agg  + (long)node * DM))[c4];
        float* dp = &hs[wave][r][c4 * 4];
        dp[0] = hv.x + av.x; dp[1] = hv.y + av.y;
        dp[2] = hv.z + av.z; dp[3] = hv.w + av.w;
    }
    __syncthreads();

    // -------- GEMM1: [16x72] @ [72x64], fully unrolled (72 WMMAs) --------
    v8f acc[4] = {};
#pragma unroll
    for (int kq = 0; kq < DM / 4; ++kq) {
        const v2f a = *(const v2f*)&hs[wave][l16][kq * 4 + 2 * half];
#pragma unroll
        for (int nt = 0; nt < 4; ++nt)
            acc[nt] = __builtin_amdgcn_wmma_f32_16x16x4_f32(
                false, a, false, bs1[(kq * 4 + nt) * 32 + lane],
                (short)0, acc[nt], false, false);
    }

    // bias + relu, write t into the (fully consumed) hs tile: A operand of GEMM2
#pragma unroll
    for (int nt = 0; nt < 4; ++nt)
#pragma unroll
        for (int j = 0; j < 8; ++j) {
            int row = j + 8 * half;             // C/D layout: VGPR j -> row j / j+8
            int col = nt * 16 + l16;
            float v = acc[nt][j] + b1s[col];
            hs[wave][row][col] = v > 0.f ? v : 0.f;
        }
    // wave-private LDS region + per-wave in-order LDS => no barrier needed

    // -------- GEMM2: [16x64] @ [64x64], fully unrolled (64 WMMAs) --------
    v8f acc2[4] = {};
#pragma unroll
    for (int kq = 0; kq < HH / 4; ++kq) {
        const v2f a = *(const v2f*)&hs[wave][l16][kq * 4 + 2 * half];
#pragma unroll
        for (int nt = 0; nt < 4; ++nt)
            acc2[nt] = __builtin_amdgcn_wmma_f32_16x16x4_f32(
                false, a, false, bs2[(kq * 4 + nt) * 32 + lane],
                (short)0, acc2[nt], false, false);
    }

    // bias + relu + store
#pragma unroll
    for (int nt = 0; nt < 4; ++nt)
#pragma unroll
        for (int j = 0; j < 8; ++j) {
            int row = j + 8 * half;
            int col = nt * 16 + l16;
            float v = acc2[nt][j] + b2s[col];
            v = v > 0.f ? v : 0.f;
            int node = node0 + row;
            if (node < n) hout[(long)node * HH + col] = v;
        }
}

// ---------------------------------------------------------------------------
// Kernel 4: global add pool  pooled[batch[n]] += h[n]   (64x64 fp32, L2-hot)
// ---------------------------------------------------------------------------
__global__ __launch_bounds__(256)
void pool_add(const float* __restrict__ h, const int* __restrict__ batch,
              float* __restrict__ pooled, int n) {
    int t = blockIdx.x * blockDim.x + threadIdx.x;
    if (t >= n * HH) return;
    int node = t >> 6;
    int c    = t & 63;
    atomicAdd(pooled + batch[node] * HH + c, h[t]);
}

// ---------------------------------------------------------------------------
// Kernel 5: final linear  out = pooled @ lin_w + lin_b   (64x32, trivial)
// ---------------------------------------------------------------------------
__global__ __launch_bounds__(256)
void final_linear(const float* __restrict__ pooled, const float* __restrict__ lw,
                  const float* __restrict__ lb, float* __restrict__ out) {
    int t = blockIdx.x * blockDim.x + threadIdx.x;
    if (t >= GG * OUTC) return;
    int g = t / OUTC;
    int o = t % OUTC;
    float acc = lb[o];
#pragma unroll 8
    for (int k = 0; k < HH; ++k)
        acc = fmaf(pooled[g * HH + k], lw[k * OUTC + o], acc);
    out[t] = acc;
}

// ---------------------------------------------------------------------------
extern "C" void kernel_launch(void* const* d_in, const int* in_sizes, int n_in,
                              void* d_out, int out_size, void* d_ws, size_t ws_size,
                              hipStream_t stream) {
    (void)in_sizes; (void)n_in; (void)out_size; (void)ws_size;
    const float* x     = (const float*)d_in[0];
    const int*   ei    = (const int*)d_in[1];
    const int*   batch = (const int*)d_in[2];
    const float* W_in  = (const float*)d_in[3];
    const float* b_in  = (const float*)d_in[4];
    const float* W_h   = (const float*)d_in[5];
    const float* b_h   = (const float*)d_in[6];
    const float* w1    = (const float*)d_in[7];
    const float* b1    = (const float*)d_in[8];
    const float* w2    = (const float*)d_in[9];
    const float* b2    = (const float*)d_in[10];
    const float* lw    = (const float*)d_in[11];
    const float* lb    = (const float*)d_in[12];

    float* h_buf  = (float*)d_ws;                      // N*64
    float* hcat   = h_buf + (size_t)NN * DD;           // N*72
    float* agg    = hcat  + (size_t)NN * DM;           // N*72
    float* pooled = agg   + (size_t)NN * DM;           // G*64

    const int* src = ei;
    const int* dst = ei + EE;

    const float* hcur = x;
    for (int l = 0; l < LL; ++l) {
        const float* W  = (l == 0) ? W_in : W_h;
        const float* bb = (l == 0) ? b_in : b_h;
        rff_concat<<<(NN + 255) / 256, 256, 0, stream>>>(hcur, W, bb, hcat, NN);
        hipMemsetAsync(agg, 0, (size_t)NN * DM * sizeof(float), stream);
        edge_agg<<<(EE + 255) / 256, 256, 0, stream>>>(src, dst, hcat, agg, EE);
        gin_mlp<<<(NN + 127) / 128, 256, 0, stream>>>(
            hcat, agg, w1 + (size_t)l * DM * HH, b1 + (size_t)l * HH,
            w2 + (size_t)l * HH * HH, b2 + (size_t)l * HH, h_buf, NN);
        hcur = h_buf;
    }
    hipMemsetAsync(pooled, 0, (size_t)GG * HH * sizeof(float), stream);
    pool_add<<<(NN * HH + 255) / 256, 256, 0, stream>>>(h_buf, batch, pooled, NN);
    final_linear<<<(GG * OUTC + 255) / 256, 256, 0, stream>>>(pooled, lw, lb,
                                                              (float*)d_out);
}